// QuantizedEmbedding_90589450207300
// MI455X (gfx1250) — compile-verified
//
#include <hip/hip_runtime.h>
#include <hip/hip_bf16.h>
#include <stdint.h>

// QuantizedEmbedding: out[t][d] = code[qw[x[t]][d]] * absmax[x[t]>>2]
// V=50304, D=1024, rows gathered = 8*2048 = 16384, BLOCKSIZE(4096) == 4 rows.

#define EMB_D      1024
#define BLOCK_THREADS 256
#define GRID_BLOCKS   2048

__global__ __launch_bounds__(BLOCK_THREADS)
void qembed_dequant_gather_kernel(const int* __restrict__ x,
                                  const int* __restrict__ qw,
                                  const float* __restrict__ absmax,
                                  const float* __restrict__ code,
                                  float* __restrict__ out,
                                  int nrows)
{
    __shared__ float s_code[256];
    __shared__ int   s_row[2][EMB_D];   // double-buffered quantized row (4KB each)

    const int tid = threadIdx.x;

    // Stage codebook into LDS (exactly 256 threads).
    s_code[tid] = code[tid];
    __syncthreads();

    // LDS byte addresses (relative to wave LDS base) for this thread's 16B slot.
    const uint32_t lds0 = (uint32_t)(uintptr_t)(&s_row[0][tid * 4]);
    const uint32_t lds1 = (uint32_t)(uintptr_t)(&s_row[1][tid * 4]);
    const unsigned long long qw_base = (unsigned long long)(uintptr_t)qw;
    const uint32_t lane_off = (uint32_t)tid * 16u;

    int t = (int)blockIdx.x;
    if (t >= nrows) return;

    // Prologue: kick off async copy of first row into buffer 0.
    int cur = __builtin_amdgcn_readfirstlane(x[t]);
    {
        uint32_t voff = (uint32_t)cur * 4096u + lane_off;   // row byte offset (int32 codes)
        asm volatile("global_load_async_to_lds_b128 %0, %1, %2"
                     :: "v"(lds0), "v"(voff), "s"(qw_base) : "memory");
    }

    int stage = 0;
    for (; t < nrows; t += (int)gridDim.x, stage ^= 1) {
        const int tn = t + (int)gridDim.x;
        int nxt = 0;
        if (tn < nrows) {
            // Prefetch next row into the other LDS buffer while we consume this one.
            nxt = __builtin_amdgcn_readfirstlane(x[tn]);
            uint32_t voff = (uint32_t)nxt * 4096u + lane_off;
            uint32_t ldsn = stage ? lds0 : lds1;
            // WAR guard: prior ds_loads from the buffer being overwritten must retire.
            asm volatile("s_wait_dscnt 0x0" ::: "memory");
            asm volatile("global_load_async_to_lds_b128 %0, %1, %2"
                         :: "v"(ldsn), "v"(voff), "s"(qw_base) : "memory");
            // Async loads complete in order: allow 1 outstanding (the prefetch),
            // guaranteeing the current row's copy has landed in LDS.
            asm volatile("s_wait_asynccnt 0x1" ::: "memory");
        } else {
            asm volatile("s_wait_asynccnt 0x0" ::: "memory");
        }

        const float scale = absmax[cur >> 2];   // one absmax block == 4 rows

        const int* rp = &s_row[stage][tid * 4];
        int q0 = rp[0];
        int q1 = rp[1];
        int q2 = rp[2];
        int q3 = rp[3];

        float4 o;
        o.x = s_code[q0 & 255] * scale;
        o.y = s_code[q1 & 255] * scale;
        o.z = s_code[q2 & 255] * scale;
        o.w = s_code[q3 & 255] * scale;

        *(float4*)(out + (size_t)t * EMB_D + (size_t)tid * 4) = o;

        cur = nxt;
    }
}

extern "C" void kernel_launch(void* const* d_in, const int* in_sizes, int n_in,
                              void* d_out, int out_size, void* d_ws, size_t ws_size,
                              hipStream_t stream) {
    // setup_inputs() order: x [8,2048] int, quant_weight [50304,1024] int,
    //                       absmax [12576] f32, code [256] f32
    const int*   x      = (const int*)d_in[0];
    const int*   qw     = (const int*)d_in[1];
    const float* absmax = (const float*)d_in[2];
    const float* code   = (const float*)d_in[3];
    float*       out    = (float*)d_out;

    const int nrows = in_sizes[0];              // 16384 gathered rows
    int grid = GRID_BLOCKS;
    if (grid > nrows) grid = nrows;

    qembed_dequant_gather_kernel<<<grid, BLOCK_THREADS, 0, stream>>>(
        x, qw, absmax, code, out, nrows);
}